// Policy_36103495090600
// MI455X (gfx1250) — compile-verified
//
#include <hip/hip_runtime.h>
#include <hip/hip_bf16.h>

typedef __attribute__((ext_vector_type(2))) float v2f;
typedef __attribute__((ext_vector_type(4))) float v4f;
typedef __attribute__((ext_vector_type(8))) float v8f;

#define BSZ   16
#define NENT  512
#define DIN   256
#define COUT  32
#define CSPA  48
#define HH    256
#define WWI   256
#define PLANE (HH * WWI)          // 65536 floats per channel plane
#define CTOT  (CSPA + COUT)       // 80 output channels

// ---------------------------------------------------------------------------
// Kernel 1: streaming copy of spatial_info into channels [0,48) and zero-fill
// of channels [48,80). Pure bandwidth: b128 non-temporal traffic.
// ---------------------------------------------------------------------------
__global__ void copy_zero_kernel(const float* __restrict__ sp,
                                 float* __restrict__ out,
                                 unsigned long long total4) {
    const unsigned long long PLANE4 = PLANE / 4;   // 16384 v4f per plane
    unsigned long long idx = (unsigned long long)blockIdx.x * blockDim.x + threadIdx.x;
    if (idx >= total4) return;

    unsigned long long plane  = idx / PLANE4;          // b*80 + c
    unsigned long long within = idx - plane * PLANE4;
    unsigned long long b      = plane / CTOT;
    unsigned long long c      = plane - b * CTOT;

    v4f v;
    if (c < CSPA) {
        const v4f* s = (const v4f*)sp;
        v = __builtin_nontemporal_load(&s[(b * CSPA + c) * PLANE4 + within]);
    } else {
        v = (v4f){0.f, 0.f, 0.f, 0.f};
    }
    __builtin_nontemporal_store(v, &((v4f*)out)[idx]);
}

// ---------------------------------------------------------------------------
// Kernel 2: proj = emb @ W + b via V_WMMA_F32_16X16X4_F32, then scatter each
// entity's 32-float channel vector to out[b, 48+c, h, w].
// One wave handles a 16-entity tile and both 16-wide column halves.
// ---------------------------------------------------------------------------
__global__ void gemm_scatter_kernel(const float* __restrict__ emb,
                                    const float* __restrict__ Wm,
                                    const float* __restrict__ bias,
                                    const int*   __restrict__ loc,
                                    float* __restrict__ out) {
    const int wave = (int)((blockIdx.x * blockDim.x + threadIdx.x) >> 5);
    const int lane = (int)(threadIdx.x & 31);
    const int row0 = wave * 16;                 // first entity row of this tile

    const int m     = lane & 15;                // row (A) / col (B) within tile
    const int khalf = (lane >> 4) * 2;          // 0 for lanes 0-15, 2 for 16-31

    // A: lane holds emb[row0+m][k+khalf .. k+khalf+1]  (aligned float2)
    const float* arow = emb + (size_t)(row0 + m) * DIN + khalf;

    v8f acc0 = {};   // columns 0..15
    v8f acc1 = {};   // columns 16..31

    for (int k = 0; k < DIN; k += 4) {
        v2f a = *(const v2f*)(arow + k);

        // B: VGPR0 = W[k+khalf][n], VGPR1 = W[k+khalf+1][n]
        const float* wp = Wm + (size_t)(k + khalf) * COUT + m;
        v2f b0, b1;
        b0.x = wp[0];          b0.y = wp[COUT];
        b1.x = wp[16];         b1.y = wp[COUT + 16];

        acc0 = __builtin_amdgcn_wmma_f32_16x16x4_f32(
            /*neg_a=*/false, a, /*neg_b=*/false, b0,
            /*c_mod=*/(short)0, acc0, /*reuse_a=*/false, /*reuse_b=*/false);
        acc1 = __builtin_amdgcn_wmma_f32_16x16x4_f32(
            /*neg_a=*/false, a, /*neg_b=*/false, b1,
            /*c_mod=*/(short)0, acc1, /*reuse_a=*/false, /*reuse_b=*/false);
    }

    // C/D layout: VGPR i, lanes 0-15 -> M=i, N=lane; lanes 16-31 -> M=i+8.
    const float bias0 = bias[m];
    const float bias1 = bias[16 + m];
    const int   b_img = row0 / NENT;            // batch index
    const int   mhi   = (lane >> 4) * 8;        // row offset 0 or 8

    const size_t out_base_img =
        ((size_t)b_img * CTOT + CSPA) * (size_t)PLANE;

#pragma unroll
    for (int i = 0; i < 8; ++i) {
        const int ment = row0 + mhi + i;        // global entity row = b*NENT + n
        const int h = loc[2 * ment + 0];
        const int w = loc[2 * ment + 1];
        const size_t base = out_base_img + (size_t)h * WWI + (size_t)w;
        out[base + (size_t)m * PLANE]        = acc0[i] + bias0;
        out[base + (size_t)(16 + m) * PLANE] = acc1[i] + bias1;
    }
}

// ---------------------------------------------------------------------------
extern "C" void kernel_launch(void* const* d_in, const int* in_sizes, int n_in,
                              void* d_out, int out_size, void* d_ws, size_t ws_size,
                              hipStream_t stream) {
    const float* sp   = (const float*)d_in[0];   // spatial_info   [16,48,256,256]
    const float* emb  = (const float*)d_in[1];   // entity_embeddings [16,512,256]
    const float* Wm   = (const float*)d_in[2];   // W [256,32]
    const float* bias = (const float*)d_in[3];   // b [32]
    const int*   loc  = (const int*)d_in[4];     // entity_location [16,512,2]
    float* out = (float*)d_out;                  // [16,80,256,256]

    const unsigned long long total4 =
        (unsigned long long)BSZ * CTOT * PLANE / 4;        // 20,971,520
    const unsigned int grid1 = (unsigned int)((total4 + 255) / 256);
    copy_zero_kernel<<<grid1, 256, 0, stream>>>(sp, out, total4);

    // 8192 entity rows / 16 rows-per-wave = 512 waves; 4 waves per block.
    gemm_scatter_kernel<<<128, 128, 0, stream>>>(emb, Wm, bias, loc, out);
}